// SpectralConv3d_74491912782351
// MI455X (gfx1250) — compile-verified
//
#include <hip/hip_runtime.h>
#include <hip/hip_bf16.h>
#include <math.h>

// ---------------------------------------------------------------------------
// 5-D DHT over (B=8, Ci=32, Z=64, Y=64, X=64):  out = (Re+Im)(FFT_5D(x)) / N
// Complex intermediate kept as two fp32 planes (re, im) in d_ws (512 MB).
// 64/32-point DFT stages run on V_WMMA_F32_16X16X4_F32; the 8-point batch
// DFT is done per-thread in registers (padding a 16x4 WMMA would waste 50%).
// ---------------------------------------------------------------------------

typedef float v2f __attribute__((ext_vector_type(2)));
typedef float v8f __attribute__((ext_vector_type(8)));

#define LDSP 65   // padded leading dim for 64-wide LDS tiles (bank-conflict free)
#define NTOT 67108864ull   // 8*32*64*64*64

__device__ __forceinline__ v8f wmma4(v2f a, v2f b, v8f c) {
  // D(16x16,f32) = A(16x4,f32) * B(4x16,f32) + C
  return __builtin_amdgcn_wmma_f32_16x16x4_f32(false, a, false, b, (short)0, c,
                                               false, false);
}

// A fragment (16x4): lanes 0-15 rows, VGPR0/1 = K0/K1; lanes 16-31 = K2/K3.
__device__ __forceinline__ v2f ldA(const float* m, int ld, int r0, int c0, int lane) {
  int r = r0 + (lane & 15);
  int c = c0 + ((lane >> 4) << 1);
  v2f a;
  a.x = m[r * ld + c];
  a.y = m[r * ld + c + 1];
  return a;
}

// B fragment (4x16): VGPR v, lanes 0-15 -> row v, lanes 16-31 -> row v+2.
__device__ __forceinline__ v2f ldB(const float* m, int ld, int r0, int c0, int lane) {
  int c = c0 + (lane & 15);
  int r = r0 + ((lane >> 4) << 1);
  v2f b;
  b.x = m[r * ld + c];
  b.y = m[(r + 1) * ld + c];
  return b;
}

// Same, but source matrix is stored transposed: element (r,c) at m[c*ld + r].
__device__ __forceinline__ v2f ldBt(const float* m, int ld, int r0, int c0, int lane) {
  int c = c0 + (lane & 15);
  int r = r0 + ((lane >> 4) << 1);
  v2f b;
  b.x = m[c * ld + r];
  b.y = m[c * ld + r + 1];
  return b;
}

// Cos[k][n] = cos(2*pi*k*n/64), Sin[k][n] = sin(2*pi*k*n/64), padded ld=65.
__device__ __forceinline__ void fill_tw64(float* sc, float* ss, int tid) {
  const float w = 6.283185307179586f / 64.0f;
  for (int i = tid; i < 4096; i += 256) {
    int k = i >> 6, n = i & 63;
    float s, c;
    __sincosf(w * (float)((k * n) & 63), &s, &c);
    sc[k * LDSP + n] = c;
    ss[k * LDSP + n] = s;
  }
}

// ---------------------------------------------------------------------------
// K1: fused x-FFT + y-FFT for one (b,c,z) slab.  Real input -> complex planes.
// ---------------------------------------------------------------------------
__global__ __launch_bounds__(256) void dht_xy_kernel(const float* __restrict__ xin,
                                                     float* __restrict__ wre,
                                                     float* __restrict__ wim) {
  __shared__ float sCos[64 * LDSP];
  __shared__ float sSin[64 * LDSP];
  __shared__ float sX  [64 * LDSP];   // input, transposed: sX[x][y]
  __shared__ float sYre[64 * LDSP];   // stage-1 result: sYre[kx][y]
  __shared__ float sYim[64 * LDSP];

  const int tid = threadIdx.x, lane = tid & 31, wave = tid >> 5;
  const int z  = blockIdx.x & 63;
  const int bc = blockIdx.x >> 6;
  const size_t base = (size_t)bc * 262144u + (size_t)z * 4096u;

  fill_tw64(sCos, sSin, tid);
  for (int i = tid; i < 4096; i += 256) {          // coalesced read, transpose to LDS
    int y = i >> 6, xx = i & 63;
    sX[xx * LDSP + y] = xin[base + i];
  }
  __syncthreads();

  // stage 1: along x, imaginary input is zero -> only 2 WMMAs per k-step
  for (int t = wave; t < 16; t += 8) {
    const int m = (t >> 2) << 4, n = (t & 3) << 4;
    v8f ac = {0.f,0.f,0.f,0.f,0.f,0.f,0.f,0.f};
    v8f as = {0.f,0.f,0.f,0.f,0.f,0.f,0.f,0.f};
#pragma unroll
    for (int k = 0; k < 16; ++k) {
      v2f aC = ldA(sCos, LDSP, m, k * 4, lane);
      v2f aS = ldA(sSin, LDSP, m, k * 4, lane);
      v2f bR = ldB(sX, LDSP, k * 4, n, lane);
      ac = wmma4(aC, bR, ac);
      as = wmma4(aS, bR, as);
    }
    const int col = n + (lane & 15);
    const int rb  = m + ((lane >> 4) << 3);
#pragma unroll
    for (int v = 0; v < 8; ++v) {
      sYre[(rb + v) * LDSP + col] = ac[v];
      sYim[(rb + v) * LDSP + col] = -as[v];
    }
  }
  __syncthreads();

  // stage 2: along y (complex).  fp32 WMMA has no A-negate -> 3rd accumulator.
  for (int t = wave; t < 16; t += 8) {
    const int m = (t >> 2) << 4, n = (t & 3) << 4;
    v8f a1 = {0.f,0.f,0.f,0.f,0.f,0.f,0.f,0.f};
    v8f a2 = {0.f,0.f,0.f,0.f,0.f,0.f,0.f,0.f};
    v8f a3 = {0.f,0.f,0.f,0.f,0.f,0.f,0.f,0.f};
#pragma unroll
    for (int k = 0; k < 16; ++k) {
      v2f aC = ldA(sCos, LDSP, m, k * 4, lane);
      v2f aS = ldA(sSin, LDSP, m, k * 4, lane);
      v2f bR = ldBt(sYre, LDSP, k * 4, n, lane);   // element (y,kx) at [kx][y]
      v2f bI = ldBt(sYim, LDSP, k * 4, n, lane);
      a1 = wmma4(aC, bR, a1);
      a1 = wmma4(aS, bI, a1);
      a2 = wmma4(aC, bI, a2);
      a3 = wmma4(aS, bR, a3);
    }
    const int col = n + (lane & 15);               // kx
    const int rb  = m + ((lane >> 4) << 3);        // ky base
#pragma unroll
    for (int v = 0; v < 8; ++v) {
      const size_t off = base + (size_t)(rb + v) * 64u + (size_t)col;
      wre[off] = a1[v];
      wim[off] = a2[v] - a3[v];
    }
  }
}

// ---------------------------------------------------------------------------
// K2: z-FFT for one (b,c,y) slab, in-place on the complex planes.
// ---------------------------------------------------------------------------
__global__ __launch_bounds__(256) void dht_z_kernel(float* __restrict__ wre,
                                                    float* __restrict__ wim) {
  __shared__ float sCos[64 * LDSP];
  __shared__ float sSin[64 * LDSP];
  __shared__ float sRe [64 * LDSP];   // sRe[z][x]
  __shared__ float sIm [64 * LDSP];

  const int tid = threadIdx.x, lane = tid & 31, wave = tid >> 5;
  const int y  = blockIdx.x & 63;
  const int bc = blockIdx.x >> 6;
  const size_t base = (size_t)bc * 262144u + (size_t)y * 64u;

  fill_tw64(sCos, sSin, tid);
  for (int i = tid; i < 4096; i += 256) {
    int zz = i >> 6, xx = i & 63;
    const size_t off = base + (size_t)zz * 4096u + (size_t)xx;
    sRe[zz * LDSP + xx] = wre[off];
    sIm[zz * LDSP + xx] = wim[off];
  }
  __syncthreads();

  for (int t = wave; t < 16; t += 8) {
    const int m = (t >> 2) << 4, n = (t & 3) << 4;
    v8f a1 = {0.f,0.f,0.f,0.f,0.f,0.f,0.f,0.f};
    v8f a2 = {0.f,0.f,0.f,0.f,0.f,0.f,0.f,0.f};
    v8f a3 = {0.f,0.f,0.f,0.f,0.f,0.f,0.f,0.f};
#pragma unroll
    for (int k = 0; k < 16; ++k) {
      v2f aC = ldA(sCos, LDSP, m, k * 4, lane);
      v2f aS = ldA(sSin, LDSP, m, k * 4, lane);
      v2f bR = ldB(sRe, LDSP, k * 4, n, lane);
      v2f bI = ldB(sIm, LDSP, k * 4, n, lane);
      a1 = wmma4(aC, bR, a1);
      a1 = wmma4(aS, bI, a1);
      a2 = wmma4(aC, bI, a2);
      a3 = wmma4(aS, bR, a3);
    }
    const int col = n + (lane & 15);               // x
    const int rb  = m + ((lane >> 4) << 3);        // kz base
#pragma unroll
    for (int v = 0; v < 8; ++v) {
      const size_t off = base + (size_t)(rb + v) * 4096u + (size_t)col;
      wre[off] = a1[v];
      wim[off] = a2[v] - a3[v];
    }
  }
}

// ---------------------------------------------------------------------------
// K3a: 32-point channel FFT for one (b,z,y) slab, in-place.
// ---------------------------------------------------------------------------
__global__ __launch_bounds__(256) void dht_c_kernel(float* __restrict__ wre,
                                                    float* __restrict__ wim) {
  __shared__ float sCos[32 * 33];
  __shared__ float sSin[32 * 33];
  __shared__ float sRe [32 * LDSP];   // sRe[ci][x]
  __shared__ float sIm [32 * LDSP];

  const int tid = threadIdx.x, lane = tid & 31, wave = tid >> 5;
  const int y = blockIdx.x & 63;
  const int z = (blockIdx.x >> 6) & 63;
  const int b = blockIdx.x >> 12;
  const size_t base = (size_t)b * 8388608u + (size_t)z * 4096u + (size_t)y * 64u;

  {
    const float w = 6.283185307179586f / 32.0f;
    for (int i = tid; i < 1024; i += 256) {
      int k = i >> 5, n = i & 31;
      float s, c;
      __sincosf(w * (float)((k * n) & 31), &s, &c);
      sCos[k * 33 + n] = c;
      sSin[k * 33 + n] = s;
    }
  }
  for (int i = tid; i < 2048; i += 256) {
    int ci = i >> 6, xx = i & 63;
    const size_t off = base + (size_t)ci * 262144u + (size_t)xx;
    sRe[ci * LDSP + xx] = wre[off];
    sIm[ci * LDSP + xx] = wim[off];
  }
  __syncthreads();

  // 8 output tiles (2 M-tiles x 4 N-tiles) == exactly one per wave
  {
    const int t = wave;
    const int m = (t >> 2) << 4, n = (t & 3) << 4;
    v8f a1 = {0.f,0.f,0.f,0.f,0.f,0.f,0.f,0.f};
    v8f a2 = {0.f,0.f,0.f,0.f,0.f,0.f,0.f,0.f};
    v8f a3 = {0.f,0.f,0.f,0.f,0.f,0.f,0.f,0.f};
#pragma unroll
    for (int k = 0; k < 8; ++k) {
      v2f aC = ldA(sCos, 33, m, k * 4, lane);
      v2f aS = ldA(sSin, 33, m, k * 4, lane);
      v2f bR = ldB(sRe, LDSP, k * 4, n, lane);
      v2f bI = ldB(sIm, LDSP, k * 4, n, lane);
      a1 = wmma4(aC, bR, a1);
      a1 = wmma4(aS, bI, a1);
      a2 = wmma4(aC, bI, a2);
      a3 = wmma4(aS, bR, a3);
    }
    const int col = n + (lane & 15);               // x
    const int rb  = m + ((lane >> 4) << 3);        // kci base
#pragma unroll
    for (int v = 0; v < 8; ++v) {
      const size_t off = base + (size_t)(rb + v) * 262144u + (size_t)col;
      wre[off] = a1[v];
      wim[off] = a2[v] - a3[v];
    }
  }
}

// ---------------------------------------------------------------------------
// K3b: 8-point batch DFT per thread + Hartley combine + 1/N scale -> d_out.
// ---------------------------------------------------------------------------
__global__ __launch_bounds__(256) void dht_b_kernel(const float* __restrict__ wre,
                                                    const float* __restrict__ wim,
                                                    float* __restrict__ out) {
  const size_t gid = (size_t)blockIdx.x * 256u + threadIdx.x;  // over (c,z,y,x): 8388608
  const size_t STR = 8388608u;
  float xr[8], xi[8], c8[8], s8[8];
#pragma unroll
  for (int b = 0; b < 8; ++b) {
    xr[b] = wre[(size_t)b * STR + gid];
    xi[b] = wim[(size_t)b * STR + gid];
  }
  const float w = 6.283185307179586f / 8.0f;
#pragma unroll
  for (int j = 0; j < 8; ++j) __sincosf(w * (float)j, &s8[j], &c8[j]);
  const float scale = 1.0f / (float)NTOT;
#pragma unroll
  for (int kb = 0; kb < 8; ++kb) {
    float yr = 0.f, yi = 0.f;
#pragma unroll
    for (int b = 0; b < 8; ++b) {
      const int j = (kb * b) & 7;
      yr += xr[b] * c8[j] + xi[b] * s8[j];
      yi += xi[b] * c8[j] - xr[b] * s8[j];
    }
    out[(size_t)kb * STR + gid] = (yr + yi) * scale;
  }
}

// ---------------------------------------------------------------------------
extern "C" void kernel_launch(void* const* d_in, const int* in_sizes, int n_in,
                              void* d_out, int out_size, void* d_ws, size_t ws_size,
                              hipStream_t stream) {
  (void)in_sizes; (void)n_in; (void)out_size; (void)ws_size;
  const float* x = (const float*)d_in[0];   // (8,32,64,64,64) fp32; w_r/w_i unused (dead code upstream)
  float* wre = (float*)d_ws;                // 256 MB re plane
  float* wim = wre + NTOT;                  // 256 MB im plane
  float* out = (float*)d_out;

  dht_xy_kernel<<<16384, 256, 0, stream>>>(x, wre, wim);   // x & y FFT (fused)
  dht_z_kernel <<<16384, 256, 0, stream>>>(wre, wim);      // z FFT
  dht_c_kernel <<<32768, 256, 0, stream>>>(wre, wim);      // channel FFT (32-pt)
  dht_b_kernel <<<32768, 256, 0, stream>>>(wre, wim, out); // batch DFT-8 + combine
}